// Advection_45526653338038
// MI455X (gfx1250) — compile-verified
//
#include <hip/hip_runtime.h>
#include <hip/hip_bf16.h>

#define GN   128
#define GV   (GN * GN * GN)        // 2,097,152 grid points
#define NWG  1024                  // main-kernel workgroups
#define TPB  256                   // 8 wave32s per workgroup

typedef float v2f __attribute__((ext_vector_type(2)));
typedef float v8f __attribute__((ext_vector_type(8)));

// Accumulate the 10 surviving grad_vars (rho, u_i*rho_u_j) at neighbor `n`
// into s[] with sign `sgn` (+1 for +shift neighbor, -1 for -shift neighbor).
__device__ __forceinline__ void accum_point(const float* __restrict__ rho,
                                            const float* __restrict__ ru0,
                                            const float* __restrict__ ru1,
                                            const float* __restrict__ ru2,
                                            unsigned n, float sgn, float* s) {
    float r  = rho[n];
    float a  = ru0[n];
    float b  = ru1[n];
    float c  = ru2[n];
    float iv = __builtin_amdgcn_rcpf(r);
    float u0 = a * iv, u1 = b * iv, u2 = c * iv;
    s[0] += sgn * r;
    s[1] += sgn * (u0 * a);  s[2] += sgn * (u0 * b);  s[3] += sgn * (u0 * c);
    s[4] += sgn * (u1 * a);  s[5] += sgn * (u1 * b);  s[6] += sgn * (u1 * c);
    s[7] += sgn * (u2 * a);  s[8] += sgn * (u2 * b);  s[9] += sgn * (u2 * c);
}

__global__ __launch_bounds__(TPB) void adv_main(const float* __restrict__ x,
                                                float* __restrict__ partials) {
    // Per-wave staging slab: 8 waves x 32 lanes x 16 vars (vars 10..15 = pad)
    __shared__ float slab[8 * 32 * 16];
    __shared__ float wgAccum[16];

    const int tid = threadIdx.x;
    if (tid < 16) wgAccum[tid] = 0.0f;

    const float* __restrict__ rho = x;
    const float* __restrict__ ru0 = x + 1u * GV;
    const float* __restrict__ ru1 = x + 2u * GV;
    const float* __restrict__ ru2 = x + 3u * GV;

    float s[10];
#pragma unroll
    for (int v = 0; v < 10; ++v) s[v] = 0.0f;

    const unsigned gsize = (unsigned)gridDim.x * TPB;      // 262144 -> 8 iters, uniform
    unsigned idx = (unsigned)blockIdx.x * TPB + tid;

    for (; idx < GV; idx += gsize) {
        // Prefetch next grid-stride iteration (speculative, L2-resident anyway)
        __builtin_prefetch(&rho[idx + gsize], 0, 1);
        __builtin_prefetch(&ru0[idx + gsize], 0, 1);
        __builtin_prefetch(&ru1[idx + gsize], 0, 1);
        __builtin_prefetch(&ru2[idx + gsize], 0, 1);

        const unsigned i1 = idx >> 14;
        const unsigned i2 = (idx >> 7) & 127u;
        const unsigned i3 = idx & 127u;
        const unsigned b23 = idx & 0x3FFFu;          // keep i2,i3
        const unsigned b13 = idx & ~(127u << 7);     // keep i1,i3
        const unsigned b12 = idx & ~127u;            // keep i1,i2

        const unsigned n1p = (((i1 + 1u)   & 127u) << 14) | b23;
        const unsigned n1m = (((i1 + 127u) & 127u) << 14) | b23;
        const unsigned n2p = b13 | (((i2 + 1u)   & 127u) << 7);
        const unsigned n2m = b13 | (((i2 + 127u) & 127u) << 7);
        const unsigned n3p = b12 | ((i3 + 1u)   & 127u);
        const unsigned n3m = b12 | ((i3 + 127u) & 127u);

        accum_point(rho, ru0, ru1, ru2, n1p, +1.0f, s);
        accum_point(rho, ru0, ru1, ru2, n1m, -1.0f, s);
        accum_point(rho, ru0, ru1, ru2, n2p, +1.0f, s);
        accum_point(rho, ru0, ru1, ru2, n2m, -1.0f, s);
        accum_point(rho, ru0, ru1, ru2, n3p, +1.0f, s);
        accum_point(rho, ru0, ru1, ru2, n3m, -1.0f, s);
    }

    // out = -sum(grads); grad = diff/(2*dx) = diff * N/2 = diff * 64
#pragma unroll
    for (int v = 0; v < 10; ++v) s[v] *= -64.0f;

    const int wave = tid >> 5;
    const int lane = tid & 31;
    float* __restrict__ myslab = &slab[wave * 32 * 16];
#pragma unroll
    for (int v = 0; v < 10; ++v) myslab[lane * 16 + v] = s[v];
#pragma unroll
    for (int v = 10; v < 16; ++v) myslab[lane * 16 + v] = 0.0f;
    __syncthreads();

    // Cross-lane reduction on the matrix unit: D = A(16x4) * ones(4x16) + C
    // A layout (ISA 7.12.2, 32-bit A 16x4): lane = M + 16*(K>=2), vgpr = K%2.
    // Rows M = variable, K = 4 source lanes per chunk => D[m,n] = sum over lanes.
    v8f acc = {};
    v2f ones; ones.x = 1.0f; ones.y = 1.0f;
    const int row = lane & 15;
    const int khi = (lane >> 4) << 1;   // 0 for lanes 0-15, 2 for lanes 16-31
#pragma unroll
    for (int chunk = 0; chunk < 8; ++chunk) {
        v2f a;
        a.x = myslab[(chunk * 4 + khi + 0) * 16 + row];
        a.y = myslab[(chunk * 4 + khi + 1) * 16 + row];
        acc = __builtin_amdgcn_wmma_f32_16x16x4_f32(
            /*neg_a=*/false, a, /*neg_b=*/false, ones,
            /*c_mod=*/(short)0, acc, /*reuse_a=*/false, /*reuse_b=*/false);
    }

    // C/D 16x16 layout: lane 0 holds (M=0..7, N=0) in acc[0..7];
    // lane 16 holds (M=8..15, N=0) -> vars 8,9 in acc[0],acc[1].
    if (lane == 0) {
#pragma unroll
        for (int m = 0; m < 8; ++m) atomicAdd(&wgAccum[m], acc[m]);
    } else if (lane == 16) {
        atomicAdd(&wgAccum[8], acc[0]);
        atomicAdd(&wgAccum[9], acc[1]);
    }
    __syncthreads();

    if (tid < 16) partials[blockIdx.x * 16 + tid] = wgAccum[tid];
}

__global__ __launch_bounds__(TPB) void adv_finalize(const float* __restrict__ partials,
                                                    float* __restrict__ out, int nwg) {
    __shared__ float red[TPB];
    const int tid = threadIdx.x;
    const int var = tid & 15;
    const int grp = tid >> 4;          // 16 groups of 16 vars
    float sum = 0.0f;
    for (int i = grp; i < nwg; i += 16) sum += partials[i * 16 + var];
    red[tid] = sum;
    __syncthreads();
#pragma unroll
    for (int st = 128; st >= 16; st >>= 1) {   // strides are multiples of 16 -> var-aligned
        if (tid < st) red[tid] += red[tid + st];
        __syncthreads();
    }
    if (tid < 10) out[tid] = red[tid];
}

extern "C" void kernel_launch(void* const* d_in, const int* in_sizes, int n_in,
                              void* d_out, int out_size, void* d_ws, size_t ws_size,
                              hipStream_t stream) {
    (void)in_sizes; (void)n_in; (void)out_size; (void)ws_size;
    const float* x = (const float*)d_in[0];   // (10, 128,128,128) fp32
    float* out = (float*)d_out;               // 10 fp32
    float* partials = (float*)d_ws;           // NWG * 16 floats = 64 KB scratch

    adv_main<<<NWG, TPB, 0, stream>>>(x, partials);
    adv_finalize<<<1, TPB, 0, stream>>>(partials, out, NWG);
}